// GAT_54546084659241
// MI455X (gfx1250) — compile-verified
//
#include <hip/hip_runtime.h>
#include <math.h>

#define F_IN   512
#define F_OUT  256
#define LALPHA 0.2f

typedef __attribute__((ext_vector_type(2))) float v2f;
typedef __attribute__((ext_vector_type(8))) float v8f;

// ---------------------------------------------------------------------------
// float atomic max via monotonic int mapping (target init value must be -inf)
// ---------------------------------------------------------------------------
__device__ __forceinline__ void atomicMaxF(float* addr, float v) {
    if (v >= 0.0f) atomicMax((int*)addr, __float_as_int(v));
    else           atomicMin((unsigned int*)addr, __float_as_uint(v));
}

// ---------------------------------------------------------------------------
// init: zero the output accumulator, set segment-max = -inf, denom = 0
// ---------------------------------------------------------------------------
__global__ void gat_init_kernel(float* __restrict__ out, float* __restrict__ mx,
                                float* __restrict__ denom, int N) {
    int i = blockIdx.x * 256 + threadIdx.x;
    if (i < N * F_OUT) out[i] = 0.0f;
    if (i < N) { mx[i] = -__builtin_inff(); denom[i] = 0.0f; }
}

// ---------------------------------------------------------------------------
// seq_fts[N,256] = x[N,512] @ W[512,256] with V_WMMA_F32_16X16X4_F32
// block = 128 threads = 4 waves; wave w computes rows [16*bx,16*bx+16) x
// cols [64w, 64w+64) as four 16x16 WMMA accumulators; K loop steps by 4.
// A frag (16x4): lane<16 holds K=k+0/k+1, lane>=16 holds K=k+2/k+3, M=lane%16
// B frag (4x16): VGPR0 = rows k+{0|2}, VGPR1 = rows k+{1|3}, N=lane%16
// C/D (16x16):   vgpr v -> row v (lanes 0-15) / row v+8 (lanes 16-31)
// ---------------------------------------------------------------------------
__global__ __launch_bounds__(128)
void gat_gemm_kernel(const float* __restrict__ x, const float* __restrict__ W,
                     float* __restrict__ seq, int N) {
    const int lane  = threadIdx.x & 31;
    const int wid   = threadIdx.x >> 5;
    const int mtile = blockIdx.x;
    const int nbase = wid * 64;

    const int  mrow = mtile * 16 + (lane & 15);
    const bool mv   = (mrow < N);
    const float* __restrict__ arow = x + (size_t)(mv ? mrow : 0) * F_IN;
    const int  ksel = (lane < 16) ? 0 : 2;
    const int  nl   = nbase + (lane & 15);

    v8f acc0 = {}, acc1 = {}, acc2 = {}, acc3 = {};

    for (int k = 0; k < F_IN; k += 4) {
        v2f a;
        float a0 = arow[k + ksel];
        float a1 = arow[k + ksel + 1];
        a.x = mv ? a0 : 0.0f;
        a.y = mv ? a1 : 0.0f;

        const float* __restrict__ b0p = W + (size_t)(k + ksel) * F_OUT + nl;
        const float* __restrict__ b1p = b0p + F_OUT;

        v2f b;
        b.x = b0p[0];  b.y = b1p[0];
        acc0 = __builtin_amdgcn_wmma_f32_16x16x4_f32(false, a, false, b, (short)0, acc0, false, false);
        b.x = b0p[16]; b.y = b1p[16];
        acc1 = __builtin_amdgcn_wmma_f32_16x16x4_f32(false, a, false, b, (short)0, acc1, false, false);
        b.x = b0p[32]; b.y = b1p[32];
        acc2 = __builtin_amdgcn_wmma_f32_16x16x4_f32(false, a, false, b, (short)0, acc2, false, false);
        b.x = b0p[48]; b.y = b1p[48];
        acc3 = __builtin_amdgcn_wmma_f32_16x16x4_f32(false, a, false, b, (short)0, acc3, false, false);
    }

    const int mbase = mtile * 16 + ((lane < 16) ? 0 : 8);
#pragma unroll
    for (int v = 0; v < 8; ++v) {
        int m = mbase + v;
        if (m < N) {
            float* __restrict__ orow = seq + (size_t)m * F_OUT + nl;
            orow[0]  = acc0[v];
            orow[16] = acc1[v];
            orow[32] = acc2[v];
            orow[48] = acc3[v];
        }
    }
}

// ---------------------------------------------------------------------------
// f1[i] = seq[i,:].a1 + b1 ; f2[i] = seq[i,:].a2 + b2   (one wave per node)
// ---------------------------------------------------------------------------
__global__ __launch_bounds__(256)
void gat_f12_kernel(const float* __restrict__ seq,
                    const float* __restrict__ a1, const float* __restrict__ a2,
                    const float* __restrict__ b1, const float* __restrict__ b2,
                    float* __restrict__ f1, float* __restrict__ f2, int N) {
    int lane = threadIdx.x & 31;
    int node = blockIdx.x * 8 + (threadIdx.x >> 5);
    if (node >= N) return;
    const float* __restrict__ row = seq + (size_t)node * F_OUT;
    float s1 = 0.0f, s2 = 0.0f;
#pragma unroll
    for (int k = 0; k < F_OUT / 32; ++k) {
        int c = lane + 32 * k;
        float v = row[c];
        s1 = fmaf(v, a1[c], s1);
        s2 = fmaf(v, a2[c], s2);
    }
#pragma unroll
    for (int o = 16; o > 0; o >>= 1) {
        s1 += __shfl_xor(s1, o, 32);
        s2 += __shfl_xor(s2, o, 32);
    }
    if (lane == 0) { f1[node] = s1 + b1[0]; f2[node] = s2 + b2[0]; }
}

// ---------------------------------------------------------------------------
// pass 1: logits + per-src segment max
// ---------------------------------------------------------------------------
__global__ void gat_logit_max_kernel(const int* __restrict__ ei,
                                     const float* __restrict__ values,
                                     const float* __restrict__ f1,
                                     const float* __restrict__ f2,
                                     float* __restrict__ logits,
                                     float* __restrict__ mx, int E) {
    int e = blockIdx.x * 256 + threadIdx.x;
    if (e >= E) return;
    int s = ei[e], d = ei[E + e];
    float l = values[e] * (f1[s] + f2[d]);
    l = (l > 0.0f) ? l : LALPHA * l;
    logits[e] = l;
    atomicMaxF(&mx[s], l);
}

// ---------------------------------------------------------------------------
// pass 2: p = exp(logit - max[src]) (in place) + per-src segment sum
// ---------------------------------------------------------------------------
__global__ void gat_exp_sum_kernel(const int* __restrict__ ei,
                                   float* __restrict__ p,
                                   const float* __restrict__ mx,
                                   float* __restrict__ denom, int E) {
    int e = blockIdx.x * 256 + threadIdx.x;
    if (e >= E) return;
    int s = ei[e];
    float v = expf(p[e] - mx[s]);
    p[e] = v;
    atomicAdd(&denom[s], v);
}

// ---------------------------------------------------------------------------
// SPMM: out[src,:] += (p/denom[src]) * seq[dst,:]  (one wave per edge;
// seq_fts and out both L2-resident -> atomics/gathers stay on-chip)
// ---------------------------------------------------------------------------
__global__ __launch_bounds__(256)
void gat_spmm_kernel(const int* __restrict__ ei, const float* __restrict__ p,
                     const float* __restrict__ denom,
                     const float* __restrict__ seq, float* __restrict__ out, int E) {
    int lane = threadIdx.x & 31;
    int e = blockIdx.x * 8 + (threadIdx.x >> 5);
    if (e >= E) return;
    int s = ei[e], d = ei[E + e];
    float coef = p[e] / denom[s];
    const float* __restrict__ srow = seq + (size_t)d * F_OUT;
    float* __restrict__ orow = out + (size_t)s * F_OUT;
#pragma unroll
    for (int k = 0; k < F_OUT / 32; ++k) {
        int c = lane + 32 * k;
        atomicAdd(&orow[c], coef * srow[c]);
    }
}

// ---------------------------------------------------------------------------
// finalize: out = elu(out + b_out)
// ---------------------------------------------------------------------------
__global__ void gat_final_kernel(float* __restrict__ out,
                                 const float* __restrict__ b_out, int total) {
    int i = blockIdx.x * 256 + threadIdx.x;
    if (i >= total) return;
    float v = out[i] + b_out[i & (F_OUT - 1)];
    out[i] = (v > 0.0f) ? v : expm1f(v);
}

// ---------------------------------------------------------------------------
extern "C" void kernel_launch(void* const* d_in, const int* in_sizes, int n_in,
                              void* d_out, int out_size, void* d_ws, size_t ws_size,
                              hipStream_t stream) {
    const float* x      = (const float*)d_in[0];
    const float* values = (const float*)d_in[1];
    const int*   ei     = (const int*)  d_in[2];
    const float* W      = (const float*)d_in[3];
    const float* a1     = (const float*)d_in[4];
    const float* b1     = (const float*)d_in[5];
    const float* a2     = (const float*)d_in[6];
    const float* b2     = (const float*)d_in[7];
    const float* b_out  = (const float*)d_in[8];

    const int N = in_sizes[0] / F_IN;
    const int E = in_sizes[1];
    float* out = (float*)d_out;

    // workspace partition (all fp32)
    float* seq   = (float*)d_ws;                 // N*F_OUT
    float* f1    = seq   + (size_t)N * F_OUT;    // N
    float* f2    = f1    + N;                    // N
    float* mx    = f2    + N;                    // N
    float* denom = mx    + N;                    // N
    float* p     = denom + N;                    // E (logits, then softmax numerators)

    const int total  = N * F_OUT;
    const int mtiles = (N + 15) / 16;

    gat_init_kernel     <<<(total + 255) / 256, 256, 0, stream>>>(out, mx, denom, N);
    gat_gemm_kernel     <<<mtiles,              128, 0, stream>>>(x, W, seq, N);
    gat_f12_kernel      <<<(N + 7) / 8,         256, 0, stream>>>(seq, a1, a2, b1, b2, f1, f2, N);
    gat_logit_max_kernel<<<(E + 255) / 256,     256, 0, stream>>>(ei, values, f1, f2, p, mx, E);
    gat_exp_sum_kernel  <<<(E + 255) / 256,     256, 0, stream>>>(ei, p, mx, denom, E);
    gat_spmm_kernel     <<<(E + 7) / 8,         256, 0, stream>>>(ei, p, denom, seq, out, E);
    gat_final_kernel    <<<(total + 255) / 256, 256, 0, stream>>>(out, b_out, total);
}